// IMG2SPHERE_71725953843242
// MI455X (gfx1250) — compile-verified
//
#include <hip/hip_runtime.h>
#include <stdint.h>

// img2sphere: projection + bilinear gather + depth-plane broadcast.
// Memory-bound: 128 MiB of output stores dominates (~5.5us @ 23.3 TB/s).

typedef float f4 __attribute__((ext_vector_type(4)));

namespace {
constexpr int   kB  = 2;
constexpr int   kS  = 8192;
constexpr int   kD  = 32;
constexpr int   kC  = 64;
constexpr int   kH  = 128;
constexpr int   kW  = 128;
constexpr int   kHW = kH * kW;
constexpr float kFX = 64.0f, kFY = 64.0f, kCX = 64.0f, kCY = 64.0f;
constexpr int   kGroupsPerBlock = 16;   // (b,s) pairs per 256-thread block
constexpr int   kThreads        = 256;  // 8 wave32
}

#if __has_builtin(__builtin_amdgcn_global_store_async_from_lds_b128)
#define IMG2SPH_ASYNC_STORE 1
// The builtin expects pointers to a 16-byte (gcc-style) int vector, with the
// destination in the global (__device__/AS1) and source in LDS (AS3) spaces.
typedef int v4i __attribute__((vector_size(16)));
typedef __attribute__((address_space(1))) v4i as1_v4i;
typedef __attribute__((address_space(3))) v4i as3_v4i;
#endif

// Bilinear 4-tap gather of one float4 (channels c4*4..c4*4+3), zero-padded
// out-of-bounds, scaled by the half-space mask `front`.
__device__ __forceinline__ f4 bilinear_tap(const float* __restrict__ img,
                                           float u, float v, int c4, float front) {
    const float x0f = floorf(u), y0f = floorf(v);
    const float wx = u - x0f,   wy = v - y0f;
    const int x0 = (int)x0f, y0 = (int)y0f;
    const int x1 = x0 + 1,   y1 = y0 + 1;
    const float mx0 = (x0 >= 0 && x0 < kW) ? 1.0f : 0.0f;
    const float mx1 = (x1 >= 0 && x1 < kW) ? 1.0f : 0.0f;
    const float my0 = (y0 >= 0 && y0 < kH) ? 1.0f : 0.0f;
    const float my1 = (y1 >= 0 && y1 < kH) ? 1.0f : 0.0f;
    const int cx0 = min(max(x0, 0), kW - 1);
    const int cx1 = min(max(x1, 0), kW - 1);
    const int cy0 = min(max(y0, 0), kH - 1);
    const int cy1 = min(max(y1, 0), kH - 1);
    // Rows are 256B contiguous; 16 consecutive lanes cover one full row (B128 each).
    const f4* r00 = (const f4*)(img + (size_t)(cy0 * kW + cx0) * kC) + c4;
    const f4* r10 = (const f4*)(img + (size_t)(cy0 * kW + cx1) * kC) + c4;
    const f4* r01 = (const f4*)(img + (size_t)(cy1 * kW + cx0) * kC) + c4;
    const f4* r11 = (const f4*)(img + (size_t)(cy1 * kW + cx1) * kC) + c4;
    const float w00 = front * (1.0f - wx) * (1.0f - wy) * mx0 * my0;
    const float w10 = front * wx          * (1.0f - wy) * mx1 * my0;
    const float w01 = front * (1.0f - wx) * wy          * mx0 * my1;
    const float w11 = front * wx          * wy          * mx1 * my1;
    const f4 a = *r00, b = *r10, c = *r01, d = *r11;
    return a * w00 + b * w10 + c * w01 + d * w11;
}

__global__ __launch_bounds__(kThreads)
void img2sphere_kernel(const float* __restrict__ atten,
                       const float* __restrict__ gamma,
                       const float* __restrict__ planes,
                       const float* __restrict__ planes_d,
                       float* __restrict__ out) {
    __shared__ float sGamma[kD];
    __shared__ f4    sRes[kGroupsPerBlock * (kC / 4)];   // 4 KiB staging for async stores

    const int tid = threadIdx.x;
    if (tid < kD) sGamma[tid] = gamma[tid];
    __syncthreads();

    float gmin = sGamma[0];
    #pragma unroll
    for (int i = 1; i < kD; ++i) gmin = fminf(gmin, sGamma[i]);

    const int     g  = tid >> 4;          // ray group within block
    const int     c4 = tid & 15;          // float4 channel column
    const int64_t p  = (int64_t)blockIdx.x * kGroupsPerBlock + g;  // (b,s) pair
    const int     b  = (int)(p / kS);

    const float dirx = planes[p * 3 + 0];
    const float diry = planes[p * 3 + 1];
    const float dirz = planes[p * 3 + 2];
    const float pd   = planes_d[p];

    const float* img   = atten + (size_t)b * kHW * kC;
    float*       gbase = out + (size_t)p * (kD * kC) + c4 * 4;

    // Fast path is valid iff z = dir.z*depth_d >= 1e-6 for every depth plane:
    // then the 1e-6 clamp never fires, front==1, and u,v are depth-invariant
    // (depth cancels in FX*P.x/z). min depth over d is gmin+pd when dir.z>0.
    const float dmin = gmin + pd;
    const bool  fast = (dirz > 0.0f) && (dmin > 0.0f) && (dirz * dmin >= 1e-6f);

    if (fast) {
        const float depth = sGamma[0] + pd;             // representative plane
        const float Px = dirx * depth, Py = diry * depth, Pz = dirz * depth;
        const float z  = fmaxf(Pz, 1e-6f);
        const float u  = kFX * Px / z + kCX;
        const float v  = kFY * Py / z + kCY;
        const f4 res = bilinear_tap(img, u, v, c4, 1.0f);
#ifdef IMG2SPH_ASYNC_STORE
        // Stage once in LDS, then stream 32 duplicated rows straight from LDS
        // via the CDNA5 async store path (ASYNCcnt; drained by S_ENDPGM).
        sRes[g * 16 + c4] = res;
        asm volatile("s_wait_dscnt 0" ::: "memory");
        as3_v4i* lsrc = (as3_v4i*)&sRes[g * 16 + c4];
        #pragma unroll
        for (int d = 0; d < kD; ++d) {
            __builtin_amdgcn_global_store_async_from_lds_b128(
                (as1_v4i*)(gbase + d * kC), lsrc, 0, 0);
        }
#else
        #pragma unroll
        for (int d = 0; d < kD; ++d)
            __builtin_nontemporal_store(res, (f4*)(gbase + d * kC));
#endif
    } else {
        // Fully general (rarely taken): per-depth projection + gather.
        for (int d = 0; d < kD; ++d) {
            const float depth = sGamma[d] + pd;
            const float Px = dirx * depth, Py = diry * depth, Pz = dirz * depth;
            const float z  = fmaxf(Pz, 1e-6f);
            const float u  = kFX * Px / z + kCX;
            const float v  = kFY * Py / z + kCY;
            const float front = (Pz > 0.0f) ? 1.0f : 0.0f;
            const f4 res = bilinear_tap(img, u, v, c4, front);
            __builtin_nontemporal_store(res, (f4*)(gbase + d * kC));
        }
    }
}

extern "C" void kernel_launch(void* const* d_in, const int* in_sizes, int n_in,
                              void* d_out, int out_size, void* d_ws, size_t ws_size,
                              hipStream_t stream) {
    const float* atten    = (const float*)d_in[0];   // [B, HW, C] f32
    const float* gamma    = (const float*)d_in[1];   // [D] f32
    const float* planes   = (const float*)d_in[2];   // [B, S, 3, 1] f32
    const float* planes_d = (const float*)d_in[3];   // [B, S] f32
    float*       out      = (float*)d_out;           // [B, S, D, C] f32

    const int nPairs = kB * kS;                      // 16384 (b,s) rays
    dim3 grid(nPairs / kGroupsPerBlock);             // 1024 blocks
    img2sphere_kernel<<<grid, kThreads, 0, stream>>>(atten, gamma, planes,
                                                     planes_d, out);
}